// TrajectoryGNN_46445776339255
// MI455X (gfx1250) — compile-verified
//
#include <hip/hip_runtime.h>
#include <hip/hip_bf16.h>
#include <stdint.h>

typedef __attribute__((ext_vector_type(2))) float v2f;
typedef __attribute__((ext_vector_type(8))) float v8f;

__device__ __forceinline__ void atomic_add_f32(float* addr, float v) {
    // native GLOBAL_ATOMIC_ADD_F32 (no CAS loop)
    unsafeAtomicAdd(addr, v);
}

// ---------------- zero workspace ----------------
__global__ void k_zero(float* __restrict__ buf, int n) {
    int i = blockIdx.x * blockDim.x + threadIdx.x;
    if (i < n) buf[i] = 0.0f;
}

// ---------------- degree count (deg[v] = in_deg[v]) ----------------
__global__ void k_degree(const long long* __restrict__ dst,
                         float* __restrict__ deg, int E) {
    int e = blockIdx.x * blockDim.x + threadIdx.x;
    if (e < E) {
        int d = (int)__builtin_nontemporal_load(&dst[e]);
        atomic_add_f32(&deg[d], 1.0f);
    }
}

// ---------------- dinv[v] = rsqrt(deg + 1)  (self-loop) ----------------
__global__ void k_dinv(float* __restrict__ deg, int N) {
    int v = blockIdx.x * blockDim.x + threadIdx.x;
    if (v < N) deg[v] = rsqrtf(deg[v] + 1.0f);
}

// ---------------- edge scatter: acc[d] += dinv[s]*dinv[d] * feat[s] (2-dim) --
__global__ void k_edge_agg(const long long* __restrict__ src,
                           const long long* __restrict__ dst,
                           const float* __restrict__ dinv,
                           const float* __restrict__ feat,
                           float* __restrict__ acc, int E) {
    int e = blockIdx.x * blockDim.x + threadIdx.x;
    if (e >= E) return;
    int s = (int)__builtin_nontemporal_load(&src[e]);
    int d = (int)__builtin_nontemporal_load(&dst[e]);
    float w = dinv[s] * dinv[d];
    float2 f = *(const float2*)(feat + 2 * s);
    atomic_add_f32(&acc[2 * d + 0], w * f.x);
    atomic_add_f32(&acc[2 * d + 1], w * f.y);
}

// ---------------- fused per-node MLP via WMMA -------------------------------
// z_full = z_agg + dinv^2 * x           (completes layer-1 aggregation)
// h      = relu(z_full @ W1^T + b1)     (16x16x4 f32 WMMA x4, K=2 padded)
// p      = h @ W2^T                     (lane FMAs + 16-lane butterfly)
// out    = b2 + dinv^2 * p              (self-loop of layer 2; edges added later)
__global__ __launch_bounds__(256) void k_mlp_wmma(
    const float* __restrict__ x, const float* __restrict__ z,
    const float* __restrict__ dinv,
    const float* __restrict__ W1, const float* __restrict__ b1,
    const float* __restrict__ W2, const float* __restrict__ b2,
    float* __restrict__ p, float* __restrict__ out, int N) {
    const int lane = threadIdx.x & 31;
    const int wave = threadIdx.x >> 5;
    const int node_base = (blockIdx.x * 8 + wave) * 16;
    if (node_base >= N) return;

    const int m  = lane & 15;   // row (A) / col (B,C,D) index
    const int hi = lane >> 4;   // 0: K=0,1 half; 1: K=2,3 half (zero pad)

    // ---- A operand: 16x4, K=2 real, K=2..3 zero ----
    v2f A = (v2f){0.0f, 0.0f};
    {
        int nv = node_base + m; if (nv >= N) nv = N - 1;
        if (hi == 0) {
            float di = dinv[nv];
            float sw = di * di;
            float2 xv = *(const float2*)(x + 2 * nv);
            float2 zv = *(const float2*)(z + 2 * nv);
            A.x = zv.x + sw * xv.x;
            A.y = zv.y + sw * xv.y;
        }
    }

    // ---- B operands (4x16 each, W1^T tiles) + bias-C + WMMA + ReLU ----
    v8f D[4];
#pragma unroll
    for (int t = 0; t < 4; ++t) {
        const int f = t * 16 + m;           // hidden feature index 0..63
        v2f B = (v2f){0.0f, 0.0f};
        if (hi == 0) {
            B.x = W1[2 * f + 0];
            B.y = W1[2 * f + 1];
        }
        const float bias = b1[f];
        v8f C = (v8f){bias, bias, bias, bias, bias, bias, bias, bias};
        D[t] = __builtin_amdgcn_wmma_f32_16x16x4_f32(
            false, A, false, B, (short)0, C, false, false);
#pragma unroll
        for (int j = 0; j < 8; ++j) D[t][j] = fmaxf(D[t][j], 0.0f);  // ReLU
    }

    // ---- second linear: contract 64 hidden -> 2 outputs ----
    float w2a[4], w2b[4];
#pragma unroll
    for (int t = 0; t < 4; ++t) {
        const int f = t * 16 + m;
        w2a[t] = W2[f];        // W2[0, f]
        w2b[t] = W2[64 + f];   // W2[1, f]
    }

#pragma unroll
    for (int j = 0; j < 8; ++j) {
        float p0 = w2a[0] * D[0][j] + w2a[1] * D[1][j] +
                   w2a[2] * D[2][j] + w2a[3] * D[3][j];
        float p1 = w2b[0] * D[0][j] + w2b[1] * D[1][j] +
                   w2b[2] * D[2][j] + w2b[3] * D[3][j];
        // reduce across the 16 lanes of each half-wave (cols 0..15)
#pragma unroll
        for (int msk = 1; msk < 16; msk <<= 1) {
            p0 += __shfl_xor(p0, msk, 32);
            p1 += __shfl_xor(p1, msk, 32);
        }
        const int node = node_base + j + 8 * hi;  // D row M = j + 8*(lane/16)
        if (m == 0 && node < N) {
            p[2 * node + 0] = p0;
            p[2 * node + 1] = p1;
            float di = dinv[node];
            float sw = di * di;
            out[2 * node + 0] = b2[0] + sw * p0;
            out[2 * node + 1] = b2[1] + sw * p1;
        }
    }
}

extern "C" void kernel_launch(void* const* d_in, const int* in_sizes, int n_in,
                              void* d_out, int out_size, void* d_ws, size_t ws_size,
                              hipStream_t stream) {
    const float*     x   = (const float*)d_in[0];
    const long long* ei  = (const long long*)d_in[1];   // int64 [2, E]
    const float*     W1  = (const float*)d_in[2];       // [64, 2]
    const float*     b1  = (const float*)d_in[3];       // [64]
    const float*     W2  = (const float*)d_in[4];       // [2, 64]
    const float*     b2  = (const float*)d_in[5];       // [2]
    float*           out = (float*)d_out;               // [N, 2]

    const int N = in_sizes[0] / 2;
    const int E = in_sizes[1] / 2;
    const long long* src = ei;
    const long long* dst = ei + E;

    // workspace layout (floats): deg/dinv [N] | z [2N] | p [2N]
    float* dinv = (float*)d_ws;
    float* z    = dinv + N;
    float* p    = dinv + 3 * (size_t)N;

    const int TB = 256;
    const int totWs = 5 * N;

    k_zero<<<(totWs + TB - 1) / TB, TB, 0, stream>>>(dinv, totWs);
    k_degree<<<(E + TB - 1) / TB, TB, 0, stream>>>(dst, dinv, E);
    k_dinv<<<(N + TB - 1) / TB, TB, 0, stream>>>(dinv, N);
    // layer-1 aggregation of raw 2-dim features
    k_edge_agg<<<(E + TB - 1) / TB, TB, 0, stream>>>(src, dst, dinv, x, z, E);
    // fused MLP (WMMA) -> p, and out initialized with bias + self-loop term
    k_mlp_wmma<<<(N + 127) / 128, TB, 0, stream>>>(x, z, dinv, W1, b1, W2, b2,
                                                   p, out, N);
    // layer-2 aggregation of 2-dim transformed features
    k_edge_agg<<<(E + TB - 1) / TB, TB, 0, stream>>>(src, dst, dinv, p, out, E);
}